// HybridSelfAttentionKernel_65481071403473
// MI455X (gfx1250) — compile-verified
//
#include <hip/hip_runtime.h>
#include <math.h>

typedef float v2f __attribute__((ext_vector_type(2)));
typedef float v8f __attribute__((ext_vector_type(8)));

#define NN 4096
#define DD 4

// ---------------------------------------------------------------------------
// Kernel 1: per row i compute q = x@R, k = x@E, then store cos/sin of half-
// angles. This removes all trig from the O(N^2) inner loop via
// cos((q-k)/2) = cos(q/2)cos(k/2) + sin(q/2)sin(k/2).
// ---------------------------------------------------------------------------
__global__ __launch_bounds__(256) void qk_precompute(
    const float* __restrict__ inp,
    const float* __restrict__ R,
    const float* __restrict__ E,
    float* __restrict__ cq, float* __restrict__ sq,
    float* __restrict__ ck, float* __restrict__ sk) {
  int i = blockIdx.x * blockDim.x + threadIdx.x;
  if (i >= NN) return;
  const float x0 = inp[i * 4 + 0];
  const float x1 = inp[i * 4 + 1];
  const float x2 = inp[i * 4 + 2];
  const float x3 = inp[i * 4 + 3];
#pragma unroll
  for (int d = 0; d < 4; ++d) {
    float q = x0 * R[0 * 4 + d] + x1 * R[1 * 4 + d] + x2 * R[2 * 4 + d] + x3 * R[3 * 4 + d];
    float k = x0 * E[0 * 4 + d] + x1 * E[1 * 4 + d] + x2 * E[2 * 4 + d] + x3 * E[3 * 4 + d];
    float qh = q * 0.5f, kh = k * 0.5f;
    cq[i * 4 + d] = __cosf(qh);
    sq[i * 4 + d] = __sinf(qh);
    ck[i * 4 + d] = __cosf(kh);
    sk[i * 4 + d] = __sinf(kh);
  }
}

// ---------------------------------------------------------------------------
// Kernel 2: fused attention. Each block owns a 16-row tile; 8 waves split the
// j-range. Per 4-j chunk, each wave builds:
//   A (16x4 f32) = exp(sim/2)           lane(h,m): K = 2h+g  (g = vgpr idx)
//   B (4x16 f32) = [ v | 1 | 0...0 ]    lane(h,n): K = 2h+g, col n
// and accumulates D += A*B with V_WMMA_F32_16X16X4_F32. Columns 0..3 of D are
// the numerator, column 4 is the softmax denominator. Deterministic serial LDS
// reduction across the 8 waves, then wave 0 normalizes and stores.
//
// B construction uses arithmetic 0/1 masks (not selects) so the v-loads stay
// unconditional: no EXEC divergence, no saveexec per iteration, and EXEC is
// guaranteed all-ones at every WMMA.
// ---------------------------------------------------------------------------
__global__ __launch_bounds__(256) void attn_main(
    const float* __restrict__ inp,
    const float* __restrict__ cq, const float* __restrict__ sq,
    const float* __restrict__ ck, const float* __restrict__ sk,
    float* __restrict__ out) {
  __shared__ float red[8 * 256];

  const int lane = threadIdx.x & 31;
  const int wave = threadIdx.x >> 5;
  const int h    = lane >> 4;   // lane half selects K pair
  const int m    = lane & 15;   // A row / B column for this lane
  const int tile = blockIdx.x;
  const int row  = tile * 16 + m;

  const float4 q_c = ((const float4*)cq)[row];
  const float4 q_s = ((const float4*)sq)[row];

  const int   vcol  = m & 3;
  const float maskV = (m < 4) ? 1.0f : 0.0f;   // cols 0..3: value matrix
  const float cOne  = (m == 4) ? 1.0f : 0.0f;  // col 4: denominator ones

  v8f acc = {};
  const int jspan = NN / 8;
  const int j0 = wave * jspan;
  const int j1 = j0 + jspan;

#pragma unroll 4
  for (int jc = j0; jc < j1; jc += 4) {
    const int ja = jc + 2 * h;      // K = 2h + 0
    const int jb = ja + 1;          // K = 2h + 1

    const float4 ca = ((const float4*)ck)[ja];
    const float4 sa = ((const float4*)sk)[ja];
    const float4 cb = ((const float4*)ck)[jb];
    const float4 sb = ((const float4*)sk)[jb];

    float pa = fabsf(fmaf(q_c.x, ca.x, q_s.x * sa.x))
             * fabsf(fmaf(q_c.y, ca.y, q_s.y * sa.y))
             * fabsf(fmaf(q_c.z, ca.z, q_s.z * sa.z))
             * fabsf(fmaf(q_c.w, ca.w, q_s.w * sa.w));
    float pb = fabsf(fmaf(q_c.x, cb.x, q_s.x * sb.x))
             * fabsf(fmaf(q_c.y, cb.y, q_s.y * sb.y))
             * fabsf(fmaf(q_c.z, cb.z, q_s.z * sb.z))
             * fabsf(fmaf(q_c.w, cb.w, q_s.w * sb.w));

    v2f a;
    a.x = __expf(pa * 0.5f);        // sim / sqrt(D), D = 4
    a.y = __expf(pb * 0.5f);

    // Unconditional loads; branchless mask-FMA keeps all lanes uniform.
    const float va = inp[ja * 4 + vcol];
    const float vb = inp[jb * 4 + vcol];
    v2f b;
    b.x = fmaf(maskV, va, cOne);
    b.y = fmaf(maskV, vb, cOne);

    acc = __builtin_amdgcn_wmma_f32_16x16x4_f32(
        /*neg_a=*/false, a, /*neg_b=*/false, b,
        /*c_mod=*/(short)0, acc, /*reuse_a=*/false, /*reuse_b=*/false);
  }

  // Stash per-wave partial D matrices, then deterministic reduction by wave 0.
#pragma unroll
  for (int r = 0; r < 8; ++r) red[wave * 256 + r * 32 + lane] = acc[r];
  __syncthreads();

  if (wave == 0) {
#pragma unroll
    for (int r = 0; r < 8; ++r) {
      float s = 0.0f;
#pragma unroll
      for (int w = 0; w < 8; ++w) s += red[w * 256 + r * 32 + lane];
      // D[r+8h][n] lives in lane (h,n); denominator is column n==4.
      const float den = __shfl(s, (h << 4) + 4, 32);
      if (m < 4) out[(tile * 16 + r + 8 * h) * 4 + m] = s / den;
    }
  }
}

// ---------------------------------------------------------------------------
extern "C" void kernel_launch(void* const* d_in, const int* in_sizes, int n_in,
                              void* d_out, int out_size, void* d_ws, size_t ws_size,
                              hipStream_t stream) {
  const float* inp = (const float*)d_in[0];   // [4096,4]
  const float* R   = (const float*)d_in[1];   // [16]
  const float* E   = (const float*)d_in[2];   // [16]
  float* out = (float*)d_out;                 // [4096,4]

  float* ws = (float*)d_ws;                   // 4 * 4096 * 4 floats = 256 KB
  float* cq = ws;
  float* sq = ws + 1 * NN * DD;
  float* ck = ws + 2 * NN * DD;
  float* sk = ws + 3 * NN * DD;

  qk_precompute<<<NN / 256, 256, 0, stream>>>(inp, R, E, cq, sq, ck, sk);
  attn_main<<<NN / 16, 256, 0, stream>>>(inp, cq, sq, ck, sk, out);
}